// NAM_55448027791691
// MI455X (gfx1250) — compile-verified
//
#include <hip/hip_runtime.h>

typedef __attribute__((ext_vector_type(2))) float v2f;
typedef __attribute__((ext_vector_type(8))) float v8f;

#define BSZ    8192
#define FDIM   64
#define HDIM   256
#define KSPLIT 4                    // grid.y slices over the feature dim
#define FPB    (FDIM / KSPLIT)      // 16 features per slice
#define WAVES  8
#define TPB    (WAVES * 32)         // 256 threads = 8 wave32
#define ROWS_PER_WG (WAVES * 16)    // each wave owns a 16-row (M) WMMA tile

#define LOG2E 1.4426950408889634f
#define LN2   0.6931471805599453f

// mish in the base-2 domain. Given hl2 = h*log2(e), t = 2^hl2 = e^h, s = t+1:
//   tanh(softplus(h)) = (s^2-1)/(s^2+1) = 1 - 2/(s^2+1)
//   mish(h) = h * (1 - 2/(s^2+1)) = ln2 * (hl2 - 2*hl2/(s^2+1))
// The ln2 factor is pre-folded into the dense_k LDS copy (WMMA B-matrix).
// Chain: pk_mul, dual_min, 2x v_exp, pk_add, pk_fma, 2x v_rcp, pk_mul, pk_fma.
__device__ __forceinline__ v2f mish2_l2(v2f hl2) {
    const v2f cmax = {22.0f, 22.0f};
    const v2f one  = {1.0f, 1.0f};
    const v2f m2   = {-2.0f, -2.0f};
    v2f hc = __builtin_elementwise_min(hl2, cmax);
    v2f t;
    t.x = __builtin_amdgcn_exp2f(hc.x);               // raw v_exp_f32 (exp2)
    t.y = __builtin_amdgcn_exp2f(hc.y);
    v2f s = t + 1.0f;                                 // pk_add, inline 1.0
    v2f d = __builtin_elementwise_fma(s, s, one);     // s^2 + 1 (pk_fma)
    v2f q;
    q.x = __builtin_amdgcn_rcpf(d.x);                 // v_rcp_f32
    q.y = __builtin_amdgcn_rcpf(d.y);
    v2f w = hl2 * q;                                  // pk_mul
    return __builtin_elementwise_fma(w, m2, hl2);     // hl2 - 2w (pk_fma)
}

__global__ __launch_bounds__(TPB) void nam_main(
    const float* __restrict__ X,        // [B,F]
    const float* __restrict__ exu_w,    // [F,H]
    const float* __restrict__ exu_b,    // [F]
    const float* __restrict__ dense_k,  // [F,H]
    float* __restrict__ partial)        // [KSPLIT, B]
{
    __shared__ float s_ew[FPB * HDIM];  // exp(exu_w)*log2e tile, 16 KB
    __shared__ float s_kd[FPB * HDIM];  // dense_k*ln2 tile,      16 KB

    const int tid = threadIdx.x;
    const int f0  = blockIdx.y * FPB;

    // One-time per-WG staging with both constants folded in.
    for (int i = tid; i < FPB * HDIM; i += TPB) {
        s_ew[i] = __expf(exu_w[f0 * HDIM + i]) * LOG2E;
        s_kd[i] = dense_k[f0 * HDIM + i] * LN2;
    }
    __syncthreads();

    const int wave = tid >> 5;
    const int lane = tid & 31;
    const int half = lane >> 4;       // 0: lanes 0-15, 1: lanes 16-31
    const int m    = lane & 15;       // A-matrix row within the 16-row tile
    const int rowBase = blockIdx.x * ROWS_PER_WG + wave * 16;
    const int b    = rowBase + m;     // global batch row for this lane
    const int koff = half * 2;        // A layout: low half holds K+0/K+1, high half K+2/K+3

    v8f acc = {};                     // f32 16x16 accumulator (8 VGPRs)

    for (int fl = 0; fl < FPB; ++fl) {
        const int f = f0 + fl;
        const float c = X[b * FDIM + f] - exu_b[f];   // (x - b_f), constant over h
        const float* pe = &s_ew[fl * HDIM + koff];
        const float* pk = &s_kd[fl * HDIM + koff];
        // Two independent mish chains per body: their pk ops fill each other's
        // post-v_exp / post-v_rcp hazard slots; two accumulator-chained wmmas.
#pragma unroll 4
        for (int hh = 0; hh < HDIM; hh += 8) {
            v2f e2a = *(const v2f*)(pe + hh);       // ds_load_2addr_b64 pairs
            v2f k2a = *(const v2f*)(pk + hh);
            v2f e2b = *(const v2f*)(pe + hh + 4);
            v2f k2b = *(const v2f*)(pk + hh + 4);
            v2f aa = mish2_l2(c * e2a);
            v2f ab = mish2_l2(c * e2b);
            acc = __builtin_amdgcn_wmma_f32_16x16x4_f32(
                false, aa, false, k2a, (short)0, acc, false, false);
            acc = __builtin_amdgcn_wmma_f32_16x16x4_f32(
                false, ab, false, k2b, (short)0, acc, false, false);
        }
    }

    // Extract column N==0: C/D layout -> VGPR j holds M=j (lanes 0-15) or M=j+8 (lanes 16-31).
    if (m == 0) {
        float* dst = &partial[blockIdx.y * BSZ + rowBase + half * 8];
#pragma unroll
        for (int j = 0; j < 8; ++j) dst[j] = acc[j];
    }
}

__global__ __launch_bounds__(256) void nam_reduce(
    const float* __restrict__ partial,  // [KSPLIT, B]
    const float* __restrict__ dense_b,  // [F]
    float* __restrict__ out)            // [B]
{
    const int b = blockIdx.x * 256 + threadIdx.x;
    float s = 0.0f;
#pragma unroll
    for (int f = 0; f < FDIM; ++f) s += dense_b[f];   // tiny, cache-resident
#pragma unroll
    for (int y = 0; y < KSPLIT; ++y) s += partial[y * BSZ + b];
    out[b] = s;
}

extern "C" void kernel_launch(void* const* d_in, const int* in_sizes, int n_in,
                              void* d_out, int out_size, void* d_ws, size_t ws_size,
                              hipStream_t stream) {
    (void)in_sizes; (void)n_in; (void)out_size; (void)ws_size;
    const float* X       = (const float*)d_in[0];
    const float* exu_w   = (const float*)d_in[1];
    const float* exu_b   = (const float*)d_in[2];
    const float* dense_k = (const float*)d_in[3];
    const float* dense_b = (const float*)d_in[4];
    float* out     = (float*)d_out;
    float* partial = (float*)d_ws;   // needs KSPLIT * BSZ * 4 = 128 KB scratch

    dim3 grid(BSZ / ROWS_PER_WG, KSPLIT);   // (64, 4) WGs x 8 waves = 2048 waves
    nam_main<<<grid, TPB, 0, stream>>>(X, exu_w, exu_b, dense_k, partial);
    nam_reduce<<<BSZ / 256, 256, 0, stream>>>(partial, dense_b, out);
}